// GCNLayer_28836410425494
// MI455X (gfx1250) — compile-verified
//
#include <hip/hip_runtime.h>

typedef __attribute__((ext_vector_type(2))) float v2f;
typedef __attribute__((ext_vector_type(8))) float v8f;

#define GCN_N_NODES 100000
#define GCN_D       64
#define GCN_N_EDGES 1600000

// ---------------------------------------------------------------------------
// Kernel 1: zero the output (atomic scatter-add needs a clean base).
// ---------------------------------------------------------------------------
__global__ __launch_bounds__(256) void gcn_zero_kernel(float4* __restrict__ out, int n4) {
    int i = blockIdx.x * blockDim.x + threadIdx.x;
    if (i < n4) out[i] = make_float4(0.f, 0.f, 0.f, 0.f);
}

// ---------------------------------------------------------------------------
// Kernel 2: support = x @ W via V_WMMA_F32_16X16X4_F32.
// One wave32 computes one 16x16 tile of `support`. K=64 -> 16 WMMA steps.
//
// Fragment layouts per CDNA5 ISA 7.12.2:
//   A (16x4 f32, 2 VGPRs): lanes 0-15 hold M=0..15; VGPR0 = {K=0 | K=2},
//     VGPR1 = {K=1 | K=3} selected by lane half.
//   B (4x16 f32, 2 VGPRs): lane (0-15) = N; VGPR r holds K=r (lanes 0-15)
//     and K=r+2 (lanes 16-31).
//   C/D (16x16 f32, 8 VGPRs): lane%16 = N; VGPR r holds M=r (lanes 0-15)
//     and M=r+8 (lanes 16-31).
// ---------------------------------------------------------------------------
__global__ __launch_bounds__(256) void gcn_gemm_wmma_kernel(
    const float* __restrict__ x,       // [N_NODES, D]
    const float* __restrict__ w,       // [D, D]
    float* __restrict__ support)       // [N_NODES, D]
{
    const int tilesN = GCN_D / 16;                       // 4
    int wave   = blockIdx.x * (blockDim.x >> 5) + (threadIdx.x >> 5);
    int tileM  = (wave / tilesN) * 16;                   // node-row tile
    int tileN  = (wave % tilesN) * 16;                   // feature-col tile
    if (tileM >= GCN_N_NODES) return;                    // 100000 % 16 == 0, exact fit

    const int lane = threadIdx.x & 31;
    const int half = lane >> 4;                          // 0: lanes 0-15, 1: lanes 16-31
    const int l    = lane & 15;

    v8f c = {};
#pragma unroll
    for (int k0 = 0; k0 < GCN_D; k0 += 4) {
        // A fragment: x[tileM + l, k0 + 2*half + {0,1}]  (K contiguous -> float2 load)
        const float* ap = x + (size_t)(tileM + l) * GCN_D + k0 + half * 2;
        v2f a; a.x = ap[0]; a.y = ap[1];
        // B fragment: w[k0 + 2*half + {0,1}, tileN + l]  (stride-D pair)
        const float* bp = w + (size_t)(k0 + half * 2) * GCN_D + tileN + l;
        v2f b; b.x = bp[0]; b.y = bp[GCN_D];
        // D = A*B + C   (8 args: neg_a, A, neg_b, B, c_mod, C, reuse_a, reuse_b)
        c = __builtin_amdgcn_wmma_f32_16x16x4_f32(false, a, false, b,
                                                  (short)0, c, false, false);
    }

    // Store C: row = tileM + r + 8*half, col = tileN + l
    float* op = support + (size_t)(tileM + half * 8) * GCN_D + tileN + l;
#pragma unroll
    for (int r = 0; r < 8; ++r)
        op[(size_t)r * GCN_D] = c[r];
}

// ---------------------------------------------------------------------------
// Kernel 3: COO SpMM scatter-add. 16 lanes per edge; each lane gathers one
// float4 (B128) of support[col] and does 4 hardware f32 atomic adds into
// out[row]. support/out both live in the 192 MB L2, so the ~820 MB of
// gather+RMW traffic stays on-die; only the edge stream hits HBM.
// ---------------------------------------------------------------------------
__global__ __launch_bounds__(256) void gcn_spmm_kernel(
    const float* __restrict__ support,     // [N_NODES, D]
    const int*   __restrict__ erow,
    const int*   __restrict__ ecol,
    const float* __restrict__ eval,
    float*       __restrict__ out)         // [N_NODES, D]
{
    long long i = (long long)blockIdx.x * blockDim.x + threadIdx.x;
    const long long total = (long long)GCN_N_EDGES * 16;
    if (i >= total) return;

    int e   = (int)(i >> 4);
    int sub = (int)(i & 15);

    int   r = erow[e];
    int   cidx = ecol[e];
    float v = eval[e];

    float4 s = ((const float4*)(support + (size_t)cidx * GCN_D))[sub];
    float* o = out + (size_t)r * GCN_D + sub * 4;

    unsafeAtomicAdd(o + 0, v * s.x);
    unsafeAtomicAdd(o + 1, v * s.y);
    unsafeAtomicAdd(o + 2, v * s.z);
    unsafeAtomicAdd(o + 3, v * s.w);
}

// ---------------------------------------------------------------------------
// Host launcher
// ---------------------------------------------------------------------------
extern "C" void kernel_launch(void* const* d_in, const int* in_sizes, int n_in,
                              void* d_out, int out_size, void* d_ws, size_t ws_size,
                              hipStream_t stream) {
    const float* x    = (const float*)d_in[0];   // [100000, 64]
    const float* w    = (const float*)d_in[1];   // [64, 64]
    const int*   erow = (const int*)d_in[2];     // [1.6M]
    const int*   ecol = (const int*)d_in[3];     // [1.6M]
    const float* ev   = (const float*)d_in[4];   // [1.6M]
    float* out = (float*)d_out;                  // [100000, 64]
    float* support = (float*)d_ws;               // 25.6 MB scratch

    // 1) out = 0
    {
        int n4 = GCN_N_NODES * GCN_D / 4;        // 1.6M float4
        gcn_zero_kernel<<<(n4 + 255) / 256, 256, 0, stream>>>((float4*)out, n4);
    }
    // 2) support = x @ W  (one wave per 16x16 tile; 8 waves / block of 256)
    {
        int tiles = (GCN_N_NODES / 16) * (GCN_D / 16);   // 6250 * 4 = 25000 waves
        int blocks = (tiles + 7) / 8;                    // 3125
        gcn_gemm_wmma_kernel<<<blocks, 256, 0, stream>>>(x, w, support);
    }
    // 3) out[row] += val * support[col]
    {
        long long threads = (long long)GCN_N_EDGES * 16; // 25.6M
        int blocks = (int)((threads + 255) / 256);       // 100000
        gcn_spmm_kernel<<<blocks, 256, 0, stream>>>(support, erow, ecol, ev, out);
    }
}